// QuantumEnhancedModel_8856222564962
// MI455X (gfx1250) — compile-verified
//
#include <hip/hip_runtime.h>

typedef __attribute__((ext_vector_type(2))) float v2f;
typedef __attribute__((ext_vector_type(8))) float v8f;

struct cplx { float re, im; };
__device__ inline cplx cmul(cplx a, cplx b){ return { a.re*b.re - a.im*b.im, a.re*b.im + a.im*b.re }; }
__device__ inline cplx cadd(cplx a, cplx b){ return { a.re+b.re, a.im+b.im }; }

__device__ inline void apply1q(cplx st[8], int wire, cplx m00, cplx m01, cplx m10, cplx m11){
  int stride = 4 >> wire;   // wire0->4, wire1->2, wire2->1 (index = w0*4 + w1*2 + w2)
  #pragma unroll
  for (int i = 0; i < 8; ++i){
    if (i & stride) continue;
    cplx a = st[i], b = st[i + stride];
    st[i]          = cadd(cmul(m00, a), cmul(m01, b));
    st[i + stride] = cadd(cmul(m10, a), cmul(m11, b));
  }
}

__device__ inline void applyCRX(cplx st[8], int ctrl, int tgt, float t){
  int cs = 4 >> ctrl, ts = 4 >> tgt;
  float c = cosf(0.5f * t), s = sinf(0.5f * t);
  #pragma unroll
  for (int i = 0; i < 8; ++i){
    if (!(i & cs)) continue;  // control bit must be 1
    if (i & ts) continue;     // visit each target pair once
    cplx a = st[i], b = st[i + ts];
    // RX: [c, -i s; -i s, c]
    cplx r0 = {  c*a.re + s*b.im,  c*a.im - s*b.re };
    cplx r1 = {  s*a.im + c*b.re, -s*a.re + c*b.im };
    st[i] = r0; st[i + ts] = r1;
  }
}

// Build the fixed 8x4 complex matrix M(theta): columns are U(theta)*{e0, e2, -i e4, -i e6}.
// ws[0..31]  = Re M (row-major 8x4), ws[32..63] = Im M.
__global__ void qsetup_kernel(const float* __restrict__ theta, float* __restrict__ M){
  int j = threadIdx.x;
  if (j >= 4) return;
  float th[8];
  #pragma unroll
  for (int i = 0; i < 8; ++i) th[i] = theta[i];
  cplx st[8];
  #pragma unroll
  for (int i = 0; i < 8; ++i) st[i] = {0.f, 0.f};
  st[j * 2] = (j < 2) ? cplx{1.f, 0.f} : cplx{0.f, -1.f};   // -i factor for s0 columns

  { float c=cosf(0.5f*th[0]), s=sinf(0.5f*th[0]); apply1q(st,0,{c,0.f},{0.f,-s},{0.f,-s},{c,0.f}); } // RX w0
  { float c=cosf(0.5f*th[1]), s=sinf(0.5f*th[1]); apply1q(st,1,{c,0.f},{-s,0.f},{s,0.f},{c,0.f}); }  // RY w1
  { float c=cosf(0.5f*th[2]), s=sinf(0.5f*th[2]); apply1q(st,2,{c,-s},{0.f,0.f},{0.f,0.f},{c,s}); }  // RZ w2
  applyCRX(st, 0, 1, th[3]);
  { float c=cosf(0.5f*th[4]), s=sinf(0.5f*th[4]); apply1q(st,2,{c,0.f},{-s,0.f},{s,0.f},{c,0.f}); }  // RY w2
  { float c=cosf(0.5f*th[5]), s=sinf(0.5f*th[5]); apply1q(st,1,{c,0.f},{0.f,-s},{0.f,-s},{c,0.f}); } // RX w1
  applyCRX(st, 1, 2, th[6]);
  { float c=cosf(0.5f*th[7]), s=sinf(0.5f*th[7]); apply1q(st,0,{c,-s},{0.f,0.f},{0.f,0.f},{c,s}); }  // RZ w0

  #pragma unroll
  for (int k = 0; k < 8; ++k){ M[k*4 + j] = st[k].re; M[32 + k*4 + j] = st[k].im; }
}

#define ROWSTRIDE 33  // 32 h-columns + 1 pad float -> conflict-free (gcd(33,64)=1)
#define MAXBLOCKS 2048

__global__ __launch_bounds__(256) void qmlp_main(
    const float* __restrict__ x, const float* __restrict__ M,
    const float* __restrict__ W1, const float* __restrict__ b1,
    const float* __restrict__ W2, const float* __restrict__ b2,
    float* __restrict__ out, int B)
{
  __shared__ float sW2[64];
  __shared__ float sB2[2];
  __shared__ float stage[256 * ROWSTRIDE];

  const int t = threadIdx.x;
  if (t >= 64 && t < 128)  sW2[t - 64] = W2[t - 64];
  if (t == 128)            { sB2[0] = b2[0]; sB2[1] = b2[1]; }

  // ---- Uniform constants -> SGPRs (s_load, used directly as VALU scalar operands) ----
  float sM[64];
  #pragma unroll
  for (int i = 0; i < 64; ++i) sM[i] = M[i];

  const int lane  = t & 31;
  const int wave  = t >> 5;
  const int n     = lane & 15;   // N (and A-row) index within tile
  const int khalf = lane >> 4;   // lane-half selects K pair per ISA f32 WMMA layout

  // ---- Preload W1 (8x32 row-major) into B-operand layout; lives in VGPRs all kernel ----
  // B 4x16 f32: VGPR r, lanes[0:15] K=r, lanes[16:31] K=r+2 ; N = lane&15
  v2f Bop[2][2];
  #pragma unroll
  for (int nt = 0; nt < 2; ++nt)
    #pragma unroll
    for (int kc = 0; kc < 2; ++kc){
      int kb = kc * 4 + khalf * 2;
      v2f tmp = { W1[(kb + 0) * 32 + nt * 16 + n],
                  W1[(kb + 1) * 32 + nt * 16 + n] };
      Bop[nt][kc] = tmp;
    }
  const float b1n[2] = { b1[n], b1[16 + n] };

  const int wbase = wave * 32;
  float* myrow = &stage[t * ROWSTRIDE];
  const int stride = (int)gridDim.x * 256;

  for (int base = (int)blockIdx.x * 256; base < B; base += stride) {
    const int sample = base + t;
    const int idx = (sample < B) ? sample : (B - 1);   // clamp: keep EXEC all-1s for WMMA

    // software prefetch of next grid-stride tile (global_prefetch_b8)
    if (base + stride < B) {
      int nidx = base + stride + t;
      if (nidx >= B) nidx = B - 1;
      __builtin_prefetch(&x[(size_t)nidx * 3], 0, 1);
    }

    // ---- encoding: x2 is a global phase -> dropped ----
    const float x0 = x[idx * 3 + 0];
    const float x1 = x[idx * 3 + 1];
    float s0, c0, s1, c1;
    __sincosf(0.5f * x0, &s0, &c0);
    __sincosf(0.5f * x1, &s1, &c1);
    const float v0 = c0*c1, v1 = c0*s1, v2 = s0*c1, v3 = s0*s1;

    // ---- probs[k] = (Re M[k].v)^2 + (Im M[k].v)^2  -> LDS staging (A-matrix source) ----
    #pragma unroll
    for (int k = 0; k < 8; ++k){
      float re = fmaf(sM[k*4+3], v3, fmaf(sM[k*4+2], v2, fmaf(sM[k*4+1], v1, sM[k*4+0]*v0)));
      float im = fmaf(sM[32+k*4+3], v3, fmaf(sM[32+k*4+2], v2, fmaf(sM[32+k*4+1], v1, sM[32+k*4+0]*v0)));
      myrow[k] = fmaf(re, re, im*im);
    }
    __syncthreads();

    // ---- Layer 1: (32x8) @ (8x32) via V_WMMA_F32_16X16X4_F32 (2 Mtiles x 2 Ntiles x 2 Kchunks) ----
    // A 16x4 f32: VGPR0 K=2*khalf, VGPR1 K=2*khalf+1 ; M-row = lane&15
    v2f A[2][2];
    #pragma unroll
    for (int mt = 0; mt < 2; ++mt)
      #pragma unroll
      for (int kc = 0; kc < 2; ++kc){
        const float* p = &stage[(wbase + mt*16 + n) * ROWSTRIDE + kc*4 + khalf*2];
        v2f tmp = { p[0], p[1] };
        A[mt][kc] = tmp;
      }

    v8f acc[2][2];
    #pragma unroll
    for (int mt = 0; mt < 2; ++mt)
      #pragma unroll
      for (int nt = 0; nt < 2; ++nt){
        v8f a = {0.f,0.f,0.f,0.f,0.f,0.f,0.f,0.f};
        a = __builtin_amdgcn_wmma_f32_16x16x4_f32(false, A[mt][0], false, Bop[nt][0],
                                                  (short)0, a, false, false);
        a = __builtin_amdgcn_wmma_f32_16x16x4_f32(false, A[mt][1], false, Bop[nt][1],
                                                  (short)0, a, false, false);
        acc[mt][nt] = a;
      }

    // ---- bias + ReLU, scatter h (D layout: VGPR i -> row i + 8*khalf, col = lane&15) ----
    #pragma unroll
    for (int mt = 0; mt < 2; ++mt)
      #pragma unroll
      for (int nt = 0; nt < 2; ++nt)
        #pragma unroll
        for (int i = 0; i < 8; ++i){
          int row = wbase + mt*16 + khalf*8 + i;
          float hv = fmaxf(acc[mt][nt][i] + b1n[nt], 0.f);
          stage[row * ROWSTRIDE + nt*16 + n] = hv;
        }
    __syncthreads();

    // ---- Layer 2: 32 -> 2 on VALU ----
    float o0 = sB2[0], o1 = sB2[1];
    #pragma unroll
    for (int nn = 0; nn < 32; ++nn){
      float hv = myrow[nn];
      o0 = fmaf(hv, sW2[nn*2 + 0], o0);
      o1 = fmaf(hv, sW2[nn*2 + 1], o1);
    }
    if (sample < B){
      v2f o = { o0, o1 };
      *(v2f*)(&out[(size_t)sample * 2]) = o;   // global_store_b64
    }
  }
}

extern "C" void kernel_launch(void* const* d_in, const int* in_sizes, int n_in,
                              void* d_out, int out_size, void* d_ws, size_t ws_size,
                              hipStream_t stream) {
  const float* x     = (const float*)d_in[0];
  const float* theta = (const float*)d_in[1];
  const float* W1    = (const float*)d_in[2];
  const float* b1    = (const float*)d_in[3];
  const float* W2    = (const float*)d_in[4];
  const float* b2    = (const float*)d_in[5];
  float* out = (float*)d_out;
  float* M   = (float*)d_ws;               // 64 floats

  const int B = in_sizes[0] / 3;
  qsetup_kernel<<<1, 32, 0, stream>>>(theta, M);
  int blocks = (B + 255) / 256;
  if (blocks > MAXBLOCKS) blocks = MAXBLOCKS;
  qmlp_main<<<blocks, 256, 0, stream>>>(x, M, W1, b1, W2, b2, out, B);
}